// Prediction_Module_60894046323229
// MI455X (gfx1250) — compile-verified
//
#include <hip/hip_runtime.h>

// MI455X / gfx1250, wave32. Grouped-GEMM for per-node-type MLP using
// V_WMMA_F32_16X16X4_F32. Nodes are bucketed by type so each 16-row tile
// shares one weight matrix (WMMA requires a single B matrix per op).

typedef __attribute__((ext_vector_type(2))) float v2f;
typedef __attribute__((ext_vector_type(8))) float v8f;

#define T_TYPES 16
#define DIN 64
#define DHID 64
#define DOUT 32
#define W0_STRIDE 72   // (2*72)%64==16 -> half-wave B rows hit disjoint LDS banks
#define W1_STRIDE 40   // (2*40)%64==16 -> same property
#define V_STRIDE 68    // 4*m bank spacing; +2-col half-wave shift stays conflict-free
#define BPT 48         // blocks per type -> 768 blocks total

// ws layout (int32): [0,16) counts | [16,32) offsets | [32,48) cursors |
//                    [64, 64+N) bucketed node ids.   (needs (64+N)*4 B)

__global__ void k_zero(int* ws) {
    if (threadIdx.x < 64) ws[threadIdx.x] = 0;
}

__global__ void k_count(const int* __restrict__ nt, int n, int* ws) {
    int i = blockIdx.x * blockDim.x + threadIdx.x;
    if (i < n) atomicAdd(&ws[nt[i] & 15], 1);
}

__global__ void k_scan(int* ws) {
    if (threadIdx.x == 0) {
        int s = 0;
        for (int t = 0; t < T_TYPES; ++t) {
            ws[16 + t] = s;   // exclusive offsets
            ws[32 + t] = 0;   // cursors
            s += ws[t];
        }
    }
}

__global__ void k_scatter(const int* __restrict__ nt, int n, int* ws) {
    int i = blockIdx.x * blockDim.x + threadIdx.x;
    if (i < n) {
        int t = nt[i] & 15;
        int p = atomicAdd(&ws[32 + t], 1);
        ws[64 + ws[16 + t] + p] = i;
    }
}

__global__ void __launch_bounds__(128)
k_mlp(const float* __restrict__ hid,
      const float* __restrict__ W0, const float* __restrict__ b0,
      const float* __restrict__ W1, const float* __restrict__ b1,
      const int* __restrict__ ws, float* __restrict__ out)
{
    __shared__ __align__(16) float sW0[DIN * W0_STRIDE];      // 18.0 KB
    __shared__ __align__(16) float sW1[DHID * W1_STRIDE];     // 10.0 KB
    __shared__ float sB0[DHID];
    __shared__ float sB1[DOUT];
    __shared__ __align__(16) float sV[4 * 16 * V_STRIDE];     // 17.0 KB (per-wave 16x64 v)
    __shared__ int   sIdx[4 * 16];

    const int t   = blockIdx.x / BPT;
    const int j   = blockIdx.x % BPT;
    const int cnt = ws[t];
    const int off = ws[16 + t];
    const int ntiles = (cnt + 15) >> 4;
    if (j * 4 >= ntiles) return;          // block-uniform early exit

    const int tid  = threadIdx.x;
    const int lane = tid & 31;
    const int wv   = tid >> 5;
    const int nn   = lane & 15;           // N / M index within half-wave
    const int kh   = lane >> 4;           // half-wave selector (K offset +2)

    // ---- cooperative weight/bias load into LDS (once per block) ----
    const float* w0g = W0 + t * (DIN * DHID);
    for (int e = tid; e < DIN * DHID; e += 128)
        sW0[(e >> 6) * W0_STRIDE + (e & 63)] = w0g[e];
    const float* w1g = W1 + t * (DHID * DOUT);
    for (int e = tid; e < DHID * DOUT; e += 128)
        sW1[(e >> 5) * W1_STRIDE + (e & 31)] = w1g[e];
    if (tid < DHID) sB0[tid] = b0[t * DHID + tid];
    if (tid < DOUT) sB1[tid] = b1[t * DOUT + tid];
    __syncthreads();

    float* sVw = sV + wv * 16 * V_STRIDE;
    int*   sIw = sIdx + wv * 16;

    for (int tile = j * 4 + wv; tile < ntiles; tile += BPT * 4) {
        // gather this tile's 16 node ids (clamp tail rows; stores masked later)
        int rib = tile * 16 + nn;
        int src = (rib < cnt) ? rib : (cnt - 1);
        int row = ws[64 + off + src];
        if (lane < 16) sIw[nn] = row;

        // ---- layer 1: A = gathered hid rows (16x64 f32) ----
        // A-fragment layout (16x4 f32): VGPR0 = K=2*kh, VGPR1 = K=2*kh+1
        v2f a[16];
        const float* hrow = hid + (long)row * DIN + 2 * kh;
#pragma unroll
        for (int k = 0; k < 16; ++k)
            a[k] = *(const v2f*)(hrow + 4 * k);

        v8f vacc[4];
#pragma unroll
        for (int nt4 = 0; nt4 < 4; ++nt4) {
            v8f acc = {};
#pragma unroll
            for (int k = 0; k < 16; ++k) {
                const int kr = 4 * k + 2 * kh;
                v2f b;
                b.x = sW0[(kr    ) * W0_STRIDE + nt4 * 16 + nn];
                b.y = sW0[(kr + 1) * W0_STRIDE + nt4 * 16 + nn];
                acc = __builtin_amdgcn_wmma_f32_16x16x4_f32(
                          false, a[k], false, b, (short)0, acc, false, false);
            }
            vacc[nt4] = acc;
        }

        // bias + stash 16x64 intermediate in per-wave LDS (D-layout -> memory)
#pragma unroll
        for (int nt4 = 0; nt4 < 4; ++nt4) {
            float bias = sB0[nt4 * 16 + nn];
#pragma unroll
            for (int r = 0; r < 8; ++r)
                sVw[(r + 8 * kh) * V_STRIDE + nt4 * 16 + nn] = vacc[nt4][r] + bias;
        }
        asm volatile("s_wait_dscnt 0x0" ::: "memory");

        // ---- layer 2: reload v as A-fragments ----
#pragma unroll
        for (int k = 0; k < 16; ++k)
            a[k] = *(const v2f*)(sVw + nn * V_STRIDE + 4 * k + 2 * kh);

        v8f oacc[2];
#pragma unroll
        for (int nt2 = 0; nt2 < 2; ++nt2) {
            v8f acc = {};
#pragma unroll
            for (int k = 0; k < 16; ++k) {
                const int kr = 4 * k + 2 * kh;
                v2f b;
                b.x = sW1[(kr    ) * W1_STRIDE + nt2 * 16 + nn];
                b.y = sW1[(kr + 1) * W1_STRIDE + nt2 * 16 + nn];
                acc = __builtin_amdgcn_wmma_f32_16x16x4_f32(
                          false, a[k], false, b, (short)0, acc, false, false);
            }
            oacc[nt2] = acc;
        }

        // bias + scatter-store (D-layout: VGPR r -> row r + 8*kh, col = nn)
#pragma unroll
        for (int nt2 = 0; nt2 < 2; ++nt2) {
            float bias = sB1[nt2 * 16 + nn];
#pragma unroll
            for (int r = 0; r < 8; ++r) {
                int m = r + 8 * kh;
                if (tile * 16 + m < cnt) {
                    int orow = sIw[m];
                    out[(long)orow * DOUT + nt2 * 16 + nn] = oacc[nt2][r] + bias;
                }
            }
        }
    }
}

extern "C" void kernel_launch(void* const* d_in, const int* in_sizes, int n_in,
                              void* d_out, int out_size, void* d_ws, size_t ws_size,
                              hipStream_t stream) {
    const float* hid = (const float*)d_in[0];
    const int*   nty = (const int*)d_in[1];
    const float* W0  = (const float*)d_in[2];
    const float* b0  = (const float*)d_in[3];
    const float* W1  = (const float*)d_in[4];
    const float* b1  = (const float*)d_in[5];
    float* out = (float*)d_out;
    int*   ws  = (int*)d_ws;
    const int n = in_sizes[1];            // N = number of nodes

    k_zero<<<1, 64, 0, stream>>>(ws);
    const int nb = (n + 255) / 256;
    k_count<<<nb, 256, 0, stream>>>(nty, n, ws);
    k_scan<<<1, 32, 0, stream>>>(ws);
    k_scatter<<<nb, 256, 0, stream>>>(nty, n, ws);
    k_mlp<<<T_TYPES * BPT, 128, 0, stream>>>(hid, W0, b0, W1, b1, ws, out);
}